// StaticCGM_16707422781820
// MI455X (gfx1250) — compile-verified
//
#include <hip/hip_runtime.h>

// Group-of-4 channel "max-out + ReLU":
//   x: (B=32, C=256, W=56, H=56) f32, cols: (G=64, GS=4) i32 (a permutation of 0..255)
//   For each (b, g, w, h): keep only the first-argmax channel of the group
//   (value passed through ReLU), zero the other three channels.
//
// Memory-bound streaming op: ~205 MB total traffic -> ~8.8 us at 23.3 TB/s.
// Strategy: b128 coalesced loads/stores with non-temporal (streaming) cache
// policy (gfx1250 TH field) since every byte is touched exactly once.

typedef float v4f __attribute__((ext_vector_type(4)));

#define B_   32
#define C_   256
#define G_   64
#define GS_  4
#define WH_  3136          // 56*56
#define NV_  784u          // WH/4 float4 per channel-plane
#define TOTAL4_ (B_ * G_ * NV_)   // 1,605,632 float4 work-items

__global__ __launch_bounds__(256) void groupmax_relu_kernel(
    const float* __restrict__ x,
    const int*   __restrict__ cols,
    float*       __restrict__ out)
{
    unsigned tid = blockIdx.x * 256u + threadIdx.x;
    if (tid >= TOTAL4_) return;

    unsigned plane = tid / NV_;          // b*G + g   (compiler: mul-shift)
    unsigned v     = tid - plane * NV_;  // float4 index within the 56x56 plane
    unsigned b     = plane >> 6;         // / G_
    unsigned g     = plane & 63u;        // % G_

    // Channel indices of this group (tiny, single cacheline, L2-resident).
    int c0 = cols[g * GS_ + 0];
    int c1 = cols[g * GS_ + 1];
    int c2 = cols[g * GS_ + 2];
    int c3 = cols[g * GS_ + 3];

    const v4f* __restrict__ xv = (const v4f*)x;
    v4f*       __restrict__ ov = (v4f*)out;

    unsigned bbase = b * (unsigned)(C_) * NV_ + v;   // (b*C + c)*NV + v, c folded below
    unsigned i0 = bbase + (unsigned)c0 * NV_;
    unsigned i1 = bbase + (unsigned)c1 * NV_;
    unsigned i2 = bbase + (unsigned)c2 * NV_;
    unsigned i3 = bbase + (unsigned)c3 * NV_;

    // Streaming loads: data is used exactly once -> non-temporal hint keeps
    // the 192 MB L2 from being churned by 205 MB of one-shot traffic.
    v4f a0 = __builtin_nontemporal_load(xv + i0);
    v4f a1 = __builtin_nontemporal_load(xv + i1);
    v4f a2 = __builtin_nontemporal_load(xv + i2);
    v4f a3 = __builtin_nontemporal_load(xv + i3);

    v4f o0 = {}, o1 = {}, o2 = {}, o3 = {};

#pragma unroll
    for (int j = 0; j < 4; ++j) {
        // First-occurrence argmax over the 4 group channels (matches jnp.argmax).
        float m = a0[j];
        int   i = 0;
        if (a1[j] > m) { m = a1[j]; i = 1; }
        if (a2[j] > m) { m = a2[j]; i = 2; }
        if (a3[j] > m) { m = a3[j]; i = 3; }
        float r = fmaxf(m, 0.0f);        // fused ReLU (v_max_num_f32)
        o0[j] = (i == 0) ? r : 0.0f;
        o1[j] = (i == 1) ? r : 0.0f;
        o2[j] = (i == 2) ? r : 0.0f;
        o3[j] = (i == 3) ? r : 0.0f;
    }

    __builtin_nontemporal_store(o0, ov + i0);
    __builtin_nontemporal_store(o1, ov + i1);
    __builtin_nontemporal_store(o2, ov + i2);
    __builtin_nontemporal_store(o3, ov + i3);
}

extern "C" void kernel_launch(void* const* d_in, const int* in_sizes, int n_in,
                              void* d_out, int out_size, void* d_ws, size_t ws_size,
                              hipStream_t stream)
{
    (void)in_sizes; (void)n_in; (void)d_ws; (void)ws_size; (void)out_size;
    const float* x    = (const float*)d_in[0];
    const int*   cols = (const int*)d_in[1];
    float*       out  = (float*)d_out;

    constexpr unsigned total  = TOTAL4_;            // 1,605,632
    constexpr unsigned blocks = (total + 255u) / 256u;  // 6272, exact
    groupmax_relu_kernel<<<blocks, 256, 0, stream>>>(x, cols, out);
}